// Hypergraph_Decoder_22454089024045
// MI455X (gfx1250) — compile-verified
//
#include <hip/hip_runtime.h>
#include <hip/hip_bf16.h>

// ---------------------------------------------------------------------------
// GCN pipeline for MI455X (gfx1250, wave32).
//  - Layer-1 feature transform uses V_WMMA_F32_16X16X4_F32 (f32-exact).
//  - Edge propagation uses float4 gathers + global_atomic_add_f32 scatters.
//  - BatchNorm (training stats) via LDS partial reduction + global atomics.
// ---------------------------------------------------------------------------

typedef float v2f __attribute__((ext_vector_type(2)));
typedef float v8f __attribute__((ext_vector_type(8)));

static inline int cdiv_i(long long a, int b) { return (int)((a + b - 1) / b); }

// ---------------- small utility kernels ----------------

__global__ void fill_kernel(float* p, float v, int n) {
    int i = blockIdx.x * blockDim.x + threadIdx.x;
    if (i < n) p[i] = v;
}

__global__ void degree_kernel(const long long* __restrict__ ei, float* __restrict__ deg, int E) {
    int e = blockIdx.x * blockDim.x + threadIdx.x;
    if (e < E) atomicAdd(&deg[(int)ei[(size_t)E + e]], 1.0f);
}

__global__ void rsqrt_kernel(float* __restrict__ d, int n) {
    int i = blockIdx.x * blockDim.x + threadIdx.x;
    if (i < n) d[i] = rsqrtf(d[i]);   // deg >= 1 always (self-loops)
}

// ---------------- layer-1 GEMM: [ROWS,216] x [216,64] via f32 WMMA ----------

__global__ __launch_bounds__(256) void gemm1_wmma(const float* __restrict__ X,
                                                  const float* __restrict__ W,
                                                  float* __restrict__ Y, int rows) {
    // W staged in LDS in K-pair-packed layout: sW[((k>>1)*64 + n)*2 + (k&1)]
    __shared__ float sW[216 * 64];
    const int tid = threadIdx.x;
    for (int s = tid; s < 216 * 64; s += 256) {
        int k = s >> 6, n = s & 63;
        sW[((k >> 1) * 64 + n) * 2 + (k & 1)] = W[s];
    }

    const int wave = tid >> 5;
    const int lane = tid & 31;
    const int l16  = lane & 15;
    const int lh   = lane >> 4;                 // 0 or 1 (K-half select)
    const int mbase = (blockIdx.x * 8 + wave) * 16;

    __syncthreads();
    if (mbase >= rows) return;                  // wave-uniform: EXEC stays all-1s

    const float* pa = X + (size_t)(mbase + l16) * 216 + lh * 2;
    __builtin_prefetch(pa, 0, 1);               // global_prefetch_b8: issued after the LDS fill barrier

    v8f c0 = {}, c1 = {}, c2 = {}, c3 = {};
    for (int kk = 0; kk < 54; ++kk) {
        v2f a = *(const v2f*)(pa + kk * 4);     // A: K = kk*4 + lh*2 + {0,1}
        const float* pbk = sW + ((size_t)(kk * 2 + lh) * 64 + l16) * 2;
        v2f b0 = *(const v2f*)(pbk +  0);       // N = l16
        v2f b1 = *(const v2f*)(pbk + 32);       // N = l16+16
        v2f b2 = *(const v2f*)(pbk + 64);       // N = l16+32
        v2f b3 = *(const v2f*)(pbk + 96);       // N = l16+48
        c0 = __builtin_amdgcn_wmma_f32_16x16x4_f32(false, a, false, b0, (short)0, c0, false, false);
        c1 = __builtin_amdgcn_wmma_f32_16x16x4_f32(false, a, false, b1, (short)0, c1, false, false);
        c2 = __builtin_amdgcn_wmma_f32_16x16x4_f32(false, a, false, b2, (short)0, c2, false, false);
        c3 = __builtin_amdgcn_wmma_f32_16x16x4_f32(false, a, false, b3, (short)0, c3, false, false);
    }
    float* py = Y + (size_t)mbase * 64;
    for (int v = 0; v < 8; ++v) {
        int m = v + lh * 8;                     // C/D layout: VGPR v -> M = v + (lane/16)*8
        py[(size_t)m * 64 +  0 + l16] = c0[v];
        py[(size_t)m * 64 + 16 + l16] = c1[v];
        py[(size_t)m * 64 + 32 + l16] = c2[v];
        py[(size_t)m * 64 + 48 + l16] = c3[v];
    }
}

// ---------------- edge propagation (gather * norm -> atomic scatter) --------

template <int TPE>   // threads per (edge,batch); F = 4*TPE channels
__global__ __launch_bounds__(256) void scatter_add_vec4(const float* __restrict__ src,
                                                        float* __restrict__ dst,
                                                        const long long* __restrict__ ei,
                                                        const float* __restrict__ dinv,
                                                        int E, int N) {
    const int F = TPE * 4;
    long long g = (long long)blockIdx.x * blockDim.x + threadIdx.x;
    long long total = (long long)(E + N) * 2 * TPE;
    if (g >= total) return;
    int cg = (int)(g % TPE);
    long long eb = g / TPE;
    int b = (int)(eb & 1);
    long long e = eb >> 1;
    int r, c;
    if (e < E) { r = (int)ei[e]; c = (int)ei[(size_t)E + e]; }
    else       { r = c = (int)(e - E); }
    float nm = dinv[r] * dinv[c];
    const float4 v = *(const float4*)(src + ((size_t)b * N + r) * F + cg * 4);
    float* p = dst + ((size_t)b * N + c) * F + cg * 4;
    atomicAdd(p + 0, v.x * nm);
    atomicAdd(p + 1, v.y * nm);
    atomicAdd(p + 2, v.z * nm);
    atomicAdd(p + 3, v.w * nm);
}

__global__ __launch_bounds__(256) void scatter_add_f3(const float* __restrict__ src,
                                                      float* __restrict__ dst,
                                                      const long long* __restrict__ ei,
                                                      const float* __restrict__ dinv,
                                                      int E, int N) {
    long long g = (long long)blockIdx.x * blockDim.x + threadIdx.x;
    long long total = (long long)(E + N) * 2;
    if (g >= total) return;
    int b = (int)(g & 1);
    long long e = g >> 1;
    int r, c;
    if (e < E) { r = (int)ei[e]; c = (int)ei[(size_t)E + e]; }
    else       { r = c = (int)(e - E); }
    float nm = dinv[r] * dinv[c];
    const float* s = src + ((size_t)b * N + r) * 3;
    float* p = dst + ((size_t)b * N + c) * 3;
    atomicAdd(p + 0, s[0] * nm);
    atomicAdd(p + 1, s[1] * nm);
    atomicAdd(p + 2, s[2] * nm);
}

// ---------------- bias (+optional pre-out) + ReLU + BN statistics -----------

template <int F>
__global__ __launch_bounds__(256) void bias_relu_stats(const float* __restrict__ agg,
                                                       const float* __restrict__ bias,
                                                       float* __restrict__ relu_out,
                                                       float* __restrict__ pre_out,
                                                       float* __restrict__ stats, int rows) {
    const int RPB = 256 / F;
    const int c = threadIdx.x % F;
    const int rslot = threadIdx.x / F;
    const float bc = bias[c];
    float s = 0.f, q = 0.f;
    for (int r0 = blockIdx.x * RPB; r0 < rows; r0 += gridDim.x * RPB) {
        int r = r0 + rslot;
        if (r < rows) {
            float v = agg[(size_t)r * F + c] + bc;
            if (pre_out) pre_out[(size_t)r * F + c] = v;
            v = fmaxf(v, 0.f);
            relu_out[(size_t)r * F + c] = v;
            s += v; q += v * v;
        }
    }
    __shared__ float ss[256], sq[256];
    ss[threadIdx.x] = s; sq[threadIdx.x] = q;
    __syncthreads();
    if (threadIdx.x < F) {
        float ts = 0.f, tq = 0.f;
        for (int j = 0; j < RPB; ++j) { ts += ss[c + j * F]; tq += sq[c + j * F]; }
        atomicAdd(&stats[c], ts);
        atomicAdd(&stats[F + c], tq);
    }
}

template <int F>
__global__ void bn_finalize(float* __restrict__ stats, const float* __restrict__ gamma,
                            const float* __restrict__ beta, float cnt) {
    int c = threadIdx.x;
    if (c < F) {
        float mean = stats[c] / cnt;
        float var  = stats[F + c] / cnt - mean * mean;   // biased var, matches jnp.var
        float sc   = gamma[c] * rsqrtf(var + 1e-5f);
        stats[2 * F + c] = sc;
        stats[3 * F + c] = beta[c] - mean * sc;
    }
}

template <int F>
__global__ __launch_bounds__(256) void bn_apply(float* __restrict__ buf,
                                                const float* __restrict__ stats, long long n) {
    long long i = (long long)blockIdx.x * blockDim.x + threadIdx.x;
    if (i < n) {
        int c = (int)(i & (F - 1));
        buf[i] = buf[i] * stats[2 * F + c] + stats[3 * F + c];
    }
}

// ---------------- small per-row GEMMs (64->8, 8->3) -------------------------

template <int FIN, int FOUT>
__global__ __launch_bounds__(256) void gemm_small(const float* __restrict__ src,
                                                  const float* __restrict__ W,
                                                  float* __restrict__ dst, int rows) {
    __shared__ float sw[FIN * FOUT];
    for (int i = threadIdx.x; i < FIN * FOUT; i += blockDim.x) sw[i] = W[i];
    __syncthreads();
    int r = blockIdx.x * blockDim.x + threadIdx.x;
    if (r >= rows) return;
    float acc[FOUT];
    for (int o = 0; o < FOUT; ++o) acc[o] = 0.f;
    const float* p = src + (size_t)r * FIN;
    for (int k = 0; k < FIN; ++k) {
        float xv = p[k];
        for (int o = 0; o < FOUT; ++o) acc[o] += xv * sw[k * FOUT + o];
    }
    float* q = dst + (size_t)r * FOUT;
    for (int o = 0; o < FOUT; ++o) q[o] = acc[o];
}

__global__ void bias_add3(float* __restrict__ out, const float* __restrict__ b3, long long n) {
    long long i = (long long)blockIdx.x * blockDim.x + threadIdx.x;
    if (i < n) out[i] += b3[(int)(i % 3)];
}

// ---------------------------------------------------------------------------

extern "C" void kernel_launch(void* const* d_in, const int* in_sizes, int n_in,
                              void* d_out, int out_size, void* d_ws, size_t ws_size,
                              hipStream_t stream) {
    const float* x   = (const float*)d_in[0];
    const float* W1  = (const float*)d_in[1];
    const float* b1  = (const float*)d_in[2];
    const float* g1  = (const float*)d_in[3];
    const float* be1 = (const float*)d_in[4];
    const float* W2  = (const float*)d_in[5];
    const float* b2  = (const float*)d_in[6];
    const float* g2  = (const float*)d_in[7];
    const float* be2 = (const float*)d_in[8];
    const float* W3  = (const float*)d_in[9];
    const float* b3  = (const float*)d_in[10];
    const long long* ei = (const long long*)d_in[11];

    const int Bb   = 2;
    const int N    = in_sizes[0] / (Bb * 216);   // 50000
    const int E    = in_sizes[11] / 2;           // 800000
    const int ROWS = Bb * N;                     // 100000

    float* ws   = (float*)d_ws;
    float* xw   = ws;                            // ROWS*64 (reused for xw2/xw3)
    float* hbuf = xw   + (size_t)ROWS * 64;      // ROWS*64: agg1 -> h (in place)
    float* x2b  = hbuf + (size_t)ROWS * 64;      // ROWS*8 : relu/BN of x1
    float* dinv = x2b  + (size_t)ROWS * 8;       // N
    float* st1  = dinv + N;                      // 4*64
    float* st2  = st1  + 4 * 64;                 // 4*8

    float* out0 = (float*)d_out;                 // [1,B,N,3]
    float* x1o  = out0 + (size_t)ROWS * 3;       // [1,B,N,8]

    // Re-zero all accumulation targets every call (atomics + graph replay).
    hipMemsetAsync(hbuf, 0, (size_t)ROWS * 64 * sizeof(float), stream);
    hipMemsetAsync(d_out, 0, (size_t)out_size * sizeof(float), stream);
    hipMemsetAsync(st1, 0, (4 * 64 + 4 * 8) * sizeof(float), stream);

    // degrees (self-loop => init 1) and D^{-1/2}
    fill_kernel<<<cdiv_i(N, 256), 256, 0, stream>>>(dinv, 1.0f, N);
    degree_kernel<<<cdiv_i(E, 256), 256, 0, stream>>>(ei, dinv, E);
    rsqrt_kernel<<<cdiv_i(N, 256), 256, 0, stream>>>(dinv, N);

    // layer 1
    gemm1_wmma<<<cdiv_i(ROWS / 16, 8), 256, 0, stream>>>(x, W1, xw, ROWS);
    long long t1 = (long long)(E + N) * 2 * 16;
    scatter_add_vec4<16><<<cdiv_i(t1, 256), 256, 0, stream>>>(xw, hbuf, ei, dinv, E, N);
    bias_relu_stats<64><<<1024, 256, 0, stream>>>(hbuf, b1, hbuf, nullptr, st1, ROWS);
    bn_finalize<64><<<1, 64, 0, stream>>>(st1, g1, be1, (float)ROWS);
    bn_apply<64><<<cdiv_i((long long)ROWS * 64, 256), 256, 0, stream>>>(hbuf, st1, (long long)ROWS * 64);

    // layer 2 (x1 goes straight to d_out; x2 = BN(relu(x1)))
    gemm_small<64, 8><<<cdiv_i(ROWS, 256), 256, 0, stream>>>(hbuf, W2, xw, ROWS);
    long long t2 = (long long)(E + N) * 2 * 2;
    scatter_add_vec4<2><<<cdiv_i(t2, 256), 256, 0, stream>>>(xw, x1o, ei, dinv, E, N);
    bias_relu_stats<8><<<1024, 256, 0, stream>>>(x1o, b2, x2b, x1o, st2, ROWS);
    bn_finalize<8><<<1, 8, 0, stream>>>(st2, g2, be2, (float)ROWS);
    bn_apply<8><<<cdiv_i((long long)ROWS * 8, 256), 256, 0, stream>>>(x2b, st2, (long long)ROWS * 8);

    // layer 3
    gemm_small<8, 3><<<cdiv_i(ROWS, 256), 256, 0, stream>>>(x2b, W3, xw, ROWS);
    long long t3 = (long long)(E + N) * 2;
    scatter_add_f3<<<cdiv_i(t3, 256), 256, 0, stream>>>(xw, out0, ei, dinv, E, N);
    bias_add3<<<cdiv_i((long long)ROWS * 3, 256), 256, 0, stream>>>(out0, b3, (long long)ROWS * 3);
}